// GCR_37778532335671
// MI455X (gfx1250) — compile-verified
//
#include <hip/hip_runtime.h>
#include <hip/hip_bf16.h>

#define N_NODES 100000
#define DIM 256

typedef float v2f __attribute__((ext_vector_type(2)));
typedef float v4f __attribute__((ext_vector_type(4)));
typedef float v8f __attribute__((ext_vector_type(8)));

// ---------------- zero fill (float4-wide) ----------------
__global__ void gcn_zero_kernel(float* __restrict__ p, int n4) {
    int i = blockIdx.x * blockDim.x + threadIdx.x;
    if (i < n4) {
        v4f z = {0.f, 0.f, 0.f, 0.f};
        *(v4f*)(p + (size_t)i * 4) = z;
    }
}

// ---------------- degree accumulation ----------------
__global__ void gcn_degree_kernel(const int* __restrict__ src, const int* __restrict__ dst,
                                  float* __restrict__ outdeg, float* __restrict__ indeg,
                                  int n_edges) {
    int e = blockIdx.x * blockDim.x + threadIdx.x;
    if (e < n_edges) {
        atomicAdd(&outdeg[src[e]], 1.0f);
        atomicAdd(&indeg[dst[e]], 1.0f);
    }
}

// ---------------- norm = clip(deg,1)^-0.5 (in place) ----------------
__global__ void gcn_norm_kernel(float* __restrict__ deg, int n) {
    int i = blockIdx.x * blockDim.x + threadIdx.x;
    if (i < n) {
        float d = fmaxf(deg[i], 1.0f);
        deg[i] = 1.0f / sqrtf(d);
    }
}

// ---------------- SpMM scatter: agg[dst] += h[src] * outnorm[src] ----------------
// one thread = (edge, 4-feature group); 64 consecutive threads share an edge
// -> coalesced float4 gather of the h row, coalesced fp32 atomics on the agg row
// (agg fits in the 192MB L2, so atomics resolve in L2).
__global__ void gcn_scatter_kernel(const float* __restrict__ h,
                                   const int* __restrict__ src, const int* __restrict__ dst,
                                   const float* __restrict__ outnorm,
                                   float* __restrict__ agg, int n_edges) {
    int tid = blockIdx.x * blockDim.x + threadIdx.x;
    int e = tid >> 6;             // DIM/4 = 64 groups per edge
    if (e >= n_edges) return;
    int f = (tid & 63) << 2;
    int s = src[e];
    int d = dst[e];
    float nrm = outnorm[s];
    v4f v = *(const v4f*)(h + (size_t)s * DIM + f);
    float* o = agg + (size_t)d * DIM + f;
    atomicAdd(o + 0, v.x * nrm);
    atomicAdd(o + 1, v.y * nrm);
    atomicAdd(o + 2, v.z * nrm);
    atomicAdd(o + 3, v.w * nrm);
}

// ---------------- out = relu( (agg * innorm[:,None]) @ W + b ) ----------------
// wave32; each wave owns a 16x64 output strip (4 column tiles) so every A
// fragment feeds 4 V_WMMA_F32_16X16X4_F32 ops. K-loop of 64 steps of 4.
// 100000 % 16 == 0 so every wave has full EXEC (WMMA requirement).
__global__ void __launch_bounds__(256)
gcn_gemm_kernel(const float* __restrict__ A, const float* __restrict__ innorm,
                const float* __restrict__ W, const float* __restrict__ bias,
                float* __restrict__ out, int nrows) {
    const int wave = threadIdx.x >> 5;
    const int lane = threadIdx.x & 31;
    const int strip = blockIdx.x * 8 + wave;         // 8 waves / block
    const int strips_per_row = DIM / 64;             // 4 strips of 64 cols
    const int trow = strip / strips_per_row;
    const int tcg  = strip % strips_per_row;
    const int row0 = trow * 16;
    if (row0 >= nrows) return;                       // wave-uniform

    const int m  = lane & 15;                        // row in tile (A) / col in tile (B)
    const int kh = (lane >> 4) << 1;                 // K sub-offset: 0 or 2
    const int row = row0 + m;
    const int colbase = tcg * 64;

    const float innrm = innorm[row];
    const float* __restrict__ Arow = A + (size_t)row * DIM;

    v8f acc[4];
    #pragma unroll
    for (int j = 0; j < 4; ++j)
        acc[j] = (v8f){0.f, 0.f, 0.f, 0.f, 0.f, 0.f, 0.f, 0.f};

    #pragma unroll 4
    for (int k = 0; k < DIM; k += 4) {
        // A fragment: lane holds (row, k+kh) and (row, k+kh+1), scaled by dst-norm
        v2f a = *(const v2f*)(Arow + k + kh);
        a = a * innrm;
        const float* Wk = W + (size_t)(k + kh) * DIM;
        #pragma unroll
        for (int j = 0; j < 4; ++j) {
            // B fragment: lane holds (k+kh, cj) and (k+kh+1, cj)
            const int cj = colbase + j * 16 + m;
            v2f b;
            b.x = Wk[cj];
            b.y = Wk[DIM + cj];
            acc[j] = __builtin_amdgcn_wmma_f32_16x16x4_f32(
                /*neg_a=*/false, a, /*neg_b=*/false, b,
                /*c_mod=*/(short)0, acc[j], /*reuse_a=*/false, /*reuse_b=*/false);
        }
    }

    // bias + relu + store (C/D layout: VGPR i -> M = i + (lane/16)*8, N = lane%16)
    const int rbase = row0 + ((lane >> 4) << 3);     // +0 or +8
    #pragma unroll
    for (int j = 0; j < 4; ++j) {
        const int c = colbase + j * 16 + (lane & 15);
        const float bc = bias[c];
        #pragma unroll
        for (int i = 0; i < 8; ++i) {
            float v = acc[j][i] + bc;
            out[(size_t)(rbase + i) * DIM + c] = fmaxf(v, 0.0f);
        }
    }
}

extern "C" void kernel_launch(void* const* d_in, const int* in_sizes, int n_in,
                              void* d_out, int out_size, void* d_ws, size_t ws_size,
                              hipStream_t stream) {
    const int*   src = (const int*)d_in[0];
    const int*   dst = (const int*)d_in[1];
    const float* X   = (const float*)d_in[2];
    const float* W1  = (const float*)d_in[3];
    const float* b1  = (const float*)d_in[4];
    const float* W2  = (const float*)d_in[5];
    const float* b2  = (const float*)d_in[6];
    float* out = (float*)d_out;

    const int n_edges = in_sizes[0];

    float* outnorm = (float*)d_ws;                   // N_NODES
    float* innorm  = outnorm + N_NODES;              // N_NODES
    float* agg     = innorm + N_NODES;               // N_NODES * DIM

    const int aggN  = N_NODES * DIM;                 // 25,600,000
    const int normN = 2 * N_NODES;                   // 200,000 (contiguous)

    // 1) degrees -> norms
    gcn_zero_kernel<<<(normN / 4 + 255) / 256, 256, 0, stream>>>(outnorm, normN / 4);
    gcn_degree_kernel<<<(n_edges + 255) / 256, 256, 0, stream>>>(src, dst, outnorm, innorm, n_edges);
    gcn_norm_kernel<<<(normN + 255) / 256, 256, 0, stream>>>(outnorm, normN);

    const int scatterThreads = n_edges * 64;
    const int gemmBlocks = (N_NODES / 16) * (DIM / 64) / 8;   // 3,125

    // 2) layer 1: agg = S(X*outnorm); d_out = relu((agg*innorm)@W1 + b1)
    gcn_zero_kernel<<<(aggN / 4 + 255) / 256, 256, 0, stream>>>(agg, aggN / 4);
    gcn_scatter_kernel<<<(scatterThreads + 255) / 256, 256, 0, stream>>>(X, src, dst, outnorm, agg, n_edges);
    gcn_gemm_kernel<<<gemmBlocks, 256, 0, stream>>>(agg, innorm, W1, b1, out, N_NODES);

    // 3) layer 2: agg = S(h1*outnorm); d_out = relu((agg*innorm)@W2 + b2)
    gcn_zero_kernel<<<(aggN / 4 + 255) / 256, 256, 0, stream>>>(agg, aggN / 4);
    gcn_scatter_kernel<<<(scatterThreads + 255) / 256, 256, 0, stream>>>(out, src, dst, outnorm, agg, n_edges);
    gcn_gemm_kernel<<<gemmBlocks, 256, 0, stream>>>(agg, innorm, W2, b2, out, N_NODES);
}